// DoubleGraphConv_84817014161895
// MI455X (gfx1250) — compile-verified
//
#include <hip/hip_runtime.h>
#include <hip/hip_bf16.h>
#include <math.h>

typedef __attribute__((ext_vector_type(2))) float v2f;
typedef __attribute__((ext_vector_type(8))) float v8f;

#define FDIM 128
#define LDS_STRIDE 132   // 128 + 4 pad: rows land on distinct banks

// ---------------------------------------------------------------------------
// Zero-fill scratch (grid-stride)
// ---------------------------------------------------------------------------
__global__ void zero_kernel(float* __restrict__ p, size_t n) {
    size_t i = (size_t)blockIdx.x * blockDim.x + threadIdx.x;
    size_t stride = (size_t)gridDim.x * blockDim.x;
    for (; i < n; i += stride) p[i] = 0.0f;
}

// ---------------------------------------------------------------------------
// Y[N,128] = X[N,128] @ W[128,128] + bias  via V_WMMA_F32_16X16X4_F32
// One block = 16 rows; 8 waves, wave w owns 16-column tile w.
// x-tile staged in LDS (padded) and shared by all 8 waves.
// ---------------------------------------------------------------------------
__global__ void __launch_bounds__(256)
gemm_bias_128(const float* __restrict__ X, const float* __restrict__ W,
              const float* __restrict__ bias, float* __restrict__ Y, int N) {
    __shared__ float xs[16 * LDS_STRIDE];

    const int rowBase = blockIdx.x * 16;

    // Cooperative load of the 16x128 x tile (clamped rows), float4 per thread.
    for (int i = threadIdx.x; i < 16 * 32; i += 256) {
        int r  = i >> 5;        // 0..15
        int c4 = i & 31;        // float4 column 0..31
        int gr = rowBase + r;
        if (gr >= N) gr = N - 1;
        float4 v = ((const float4*)(X + (size_t)gr * FDIM))[c4];
        float* dst = &xs[r * LDS_STRIDE + c4 * 4];
        dst[0] = v.x; dst[1] = v.y; dst[2] = v.z; dst[3] = v.w;
    }
    __syncthreads();

    const int lane    = threadIdx.x & 31;
    const int wave    = threadIdx.x >> 5;
    const int colBase = wave * 16;
    const int col     = colBase + (lane & 15);
    const int khalf   = (lane >> 4) << 1;   // lanes 0-15 -> K+0/1, lanes 16-31 -> K+2/3
    const int arow    = lane & 15;

    v8f c = {};
    for (int k = 0; k < FDIM; k += 4) {
        v2f a, b;
        // A fragment: 16x4 f32, M = lane%16, K split across lane halves
        a.x = xs[arow * LDS_STRIDE + k + khalf];
        a.y = xs[arow * LDS_STRIDE + k + khalf + 1];
        // B fragment: 4x16 f32, N = lane%16, K split across lane halves
        b.x = W[(size_t)(k + khalf)     * FDIM + col];
        b.y = W[(size_t)(k + khalf + 1) * FDIM + col];
        c = __builtin_amdgcn_wmma_f32_16x16x4_f32(
                /*neg_a=*/false, a, /*neg_b=*/false, b,
                /*c_mod=*/(short)0, c, /*reuse_a=*/false, /*reuse_b=*/false);
    }

    const float bv = bias[col];
    const int mhalf = (lane >> 4) << 3;     // lanes 16-31 hold rows M=8..15
#pragma unroll
    for (int r = 0; r < 8; ++r) {
        int row = rowBase + r + mhalf;
        if (row < N) Y[(size_t)row * FDIM + col] = c[r] + bv;
    }
}

// ---------------------------------------------------------------------------
// Edge scatter: one wave per edge (grid-stride over edges).
// acc[dst] += ew * H[src]; optional cnt[dst] += 1.  L2-resident f32 atomics.
// ---------------------------------------------------------------------------
__global__ void __launch_bounds__(256)
scatter_mean(const float* __restrict__ H, const long long* __restrict__ ei,
             const float* __restrict__ ew, float* __restrict__ acc,
             float* __restrict__ cnt, int E, int doCount) {
    const int lane = threadIdx.x & 31;
    const int wid  = (int)(((size_t)blockIdx.x * blockDim.x + threadIdx.x) >> 5);
    const int nw   = (int)(((size_t)gridDim.x * blockDim.x) >> 5);

    for (int e = wid; e < E; e += nw) {
        long long s = ei[e];
        long long d = ei[(size_t)E + e];
        float w = ew[e];
        float4 v = ((const float4*)(H + (size_t)s * FDIM))[lane];
        float* ap = acc + (size_t)d * FDIM + lane * 4;
        atomicAdd(ap + 0, v.x * w);
        atomicAdd(ap + 1, v.y * w);
        atomicAdd(ap + 2, v.z * w);
        atomicAdd(ap + 3, v.w * w);
        if (doCount && lane == 0) atomicAdd(&cnt[d], 1.0f);
    }
}

// ---------------------------------------------------------------------------
// x1 = acc / max(cnt, 1)
// ---------------------------------------------------------------------------
__global__ void divide_kernel(const float* __restrict__ acc,
                              const float* __restrict__ cnt,
                              float* __restrict__ x1, int N) {
    int i = blockIdx.x * blockDim.x + threadIdx.x;
    int total = N * FDIM;
    if (i >= total) return;
    float c = cnt[i >> 7];
    if (c < 1.0f) c = 1.0f;
    x1[i] = acc[i] / c;
}

// ---------------------------------------------------------------------------
// Fused epilogue, one wave per node:
//   x2 = (acc2 + h2) / (cnt + 1)        (self-loop w=1 folded in)
//   v  = x2 + x1                         (residual)
//   out = gelu(layernorm(v))             (exact erf GELU)
// Wave32 __shfl_xor reduction for mean/var over 128 features (4 per lane).
// ---------------------------------------------------------------------------
__global__ void __launch_bounds__(256)
final_kernel(const float* __restrict__ acc2, const float* __restrict__ h2,
             const float* __restrict__ x1, const float* __restrict__ cnt,
             const float* __restrict__ gamma, const float* __restrict__ beta,
             float* __restrict__ out, int N) {
    const int lane = threadIdx.x & 31;
    const int node = (int)(((size_t)blockIdx.x * blockDim.x + threadIdx.x) >> 5);
    if (node >= N) return;

    const size_t base = (size_t)node * FDIM;
    float4 a = ((const float4*)(acc2 + base))[lane];
    float4 h = ((const float4*)(h2   + base))[lane];
    float4 r = ((const float4*)(x1   + base))[lane];
    const float inv = 1.0f / (cnt[node] + 1.0f);

    float v0 = (a.x + h.x) * inv + r.x;
    float v1 = (a.y + h.y) * inv + r.y;
    float v2 = (a.z + h.z) * inv + r.z;
    float v3 = (a.w + h.w) * inv + r.w;

    float s  = v0 + v1 + v2 + v3;
    float s2 = v0 * v0 + v1 * v1 + v2 * v2 + v3 * v3;
#pragma unroll
    for (int m = 16; m >= 1; m >>= 1) {
        s  += __shfl_xor(s,  m, 32);
        s2 += __shfl_xor(s2, m, 32);
    }
    const float mu  = s * (1.0f / FDIM);
    const float var = s2 * (1.0f / FDIM) - mu * mu;
    const float rs  = rsqrtf(var + 1e-5f);

    float4 g = ((const float4*)gamma)[lane];
    float4 b = ((const float4*)beta)[lane];

    float4 o;
    float t;
    t = (v0 - mu) * rs * g.x + b.x; o.x = 0.5f * t * (1.0f + erff(t * 0.70710678118654752f));
    t = (v1 - mu) * rs * g.y + b.y; o.y = 0.5f * t * (1.0f + erff(t * 0.70710678118654752f));
    t = (v2 - mu) * rs * g.z + b.z; o.z = 0.5f * t * (1.0f + erff(t * 0.70710678118654752f));
    t = (v3 - mu) * rs * g.w + b.w; o.w = 0.5f * t * (1.0f + erff(t * 0.70710678118654752f));

    ((float4*)(out + base))[lane] = o;
}

// ---------------------------------------------------------------------------
// Orchestration
// ---------------------------------------------------------------------------
extern "C" void kernel_launch(void* const* d_in, const int* in_sizes, int n_in,
                              void* d_out, int out_size, void* d_ws, size_t ws_size,
                              hipStream_t stream) {
    const float*     x     = (const float*)d_in[0];
    const long long* ei    = (const long long*)d_in[1];   // int64 [2,E]
    const float*     ew    = (const float*)d_in[2];
    const float*     Wl1   = (const float*)d_in[3];
    const float*     bl1   = (const float*)d_in[4];
    const float*     Wl2   = (const float*)d_in[5];
    const float*     bl2   = (const float*)d_in[6];
    const float*     gamma = (const float*)d_in[7];
    const float*     beta  = (const float*)d_in[8];
    float*           out   = (float*)d_out;

    const int N = in_sizes[0] / FDIM;
    const int E = in_sizes[2];
    const size_t ND = (size_t)N * FDIM;

    // Workspace layout (floats): h | x1 | acc | acc2 | cnt
    float* ws   = (float*)d_ws;
    float* h    = ws;              // reused as h2 for layer 2
    float* x1   = ws + ND;
    float* acc  = ws + 2 * ND;
    float* acc2 = ws + 3 * ND;
    float* cnt  = ws + 4 * ND;

    // Zero acc, acc2, cnt (contiguous: 2*ND + N floats) — every call.
    zero_kernel<<<1024, 256, 0, stream>>>(acc, 2 * ND + (size_t)N);

    const int gemmBlocks = (N + 15) / 16;

    // Layer 1: h = x @ Wl1 + bl1
    gemm_bias_128<<<gemmBlocks, 256, 0, stream>>>(x, Wl1, bl1, h, N);
    // scatter + degree count
    scatter_mean<<<4096, 256, 0, stream>>>(h, ei, ew, acc, cnt, E, 1);
    // x1 = acc / max(cnt,1)
    {
        int total = N * FDIM;
        divide_kernel<<<(total + 255) / 256, 256, 0, stream>>>(acc, cnt, x1, N);
    }

    // Layer 2: h2 = x1 @ Wl2 + bl2 (reuse h buffer)
    gemm_bias_128<<<gemmBlocks, 256, 0, stream>>>(x1, Wl2, bl2, h, N);
    scatter_mean<<<4096, 256, 0, stream>>>(h, ei, ew, acc2, cnt, E, 0);

    // Fused self-loop mean + residual + LayerNorm + GELU
    final_kernel<<<(N + 7) / 8, 256, 0, stream>>>(acc2, h, x1, cnt, gamma, beta, out, N);
}